// LinearDynamicalSS_2465311228007
// MI455X (gfx1250) — compile-verified
//
#include <hip/hip_runtime.h>
#include <hip/hip_bf16.h>

typedef __attribute__((ext_vector_type(2))) float v2f;
typedef __attribute__((ext_vector_type(8))) float v8f;

#define HAS_WMMA_F32X4 (__has_builtin(__builtin_amdgcn_wmma_f32_16x16x4_f32))

namespace {
constexpr int B_ = 16, T_ = 8192, NU_ = 8, NX_ = 128, NO_ = 4;
constexpr int NC_ = 32;             // chunks along T
constexpr int L_  = T_ / NC_;       // 256 steps per chunk
constexpr int XS_ = 132;            // padded LDS row stride: bank = (4*row+col)%64 -> conflict-free

// workspace layout (float offsets)
constexpr int OFF_ARE = 0;                        // [NX]
constexpr int OFF_AIM = OFF_ARE + NX_;            // [NX]
constexpr int OFF_BS  = OFF_AIM + NX_;            // [NX*NU]  gamma_log * b_re
constexpr int OFF_PIN = OFF_BS + NX_ * NU_;       // [NX*NO]  Re(pinv)
constexpr int OFF_X0  = OFF_PIN + NX_ * NO_;      // [B*NX]
constexpr int OFF_FRE = OFF_X0 + B_ * NX_;        // [B*NC*NX] chunk-final state (re)
constexpr int OFF_FIM = OFF_FRE + B_ * NC_ * NX_;
constexpr int OFF_CRE = OFF_FIM + B_ * NC_ * NX_; // [B*NC*NX] carry-in per chunk (re)
constexpr int OFF_CIM = OFF_CRE + B_ * NC_ * NX_;

// dynamic LDS layout for the scan kernels (float offsets)
constexpr int SM_UT  = 0;                 // [16*NU]  = 128
constexpr int SM_XRE = SM_UT + 16 * NU_;  // [16*XS]
constexpr int SM_XIM = SM_XRE + 16 * XS_; // [16*XS]
constexpr int SM_BRE = SM_XIM + 16 * XS_; // [16*XS]  zero-padded C_re  (16 N-rows)
constexpr int SM_BIM = SM_BRE + 16 * XS_; // [16*XS]  zero-padded -C_im
constexpr int SM_D   = SM_BIM + 16 * XS_; // [NO*NU]  = 32
constexpr int SM_PC  = SM_D + NO_ * NU_;  // [4*256]  per-wave partial C fragments
constexpr int SM_TOT3 = SM_PC + 4 * 256;  // phase-3 total floats
constexpr int SM_TOT1 = 16 * NU_;         // phase-1 total floats (uT only)
}  // namespace

// ---------------------------------------------------------------------------
// Setup: derived params, 4x4 complex pinv (A+ = A^H (A A^H)^-1), x0.
// One block of 128 threads (thread = state channel s).
// ---------------------------------------------------------------------------
__global__ void ldss_setup(const float* __restrict__ y0,
                           const float* __restrict__ nu_log,
                           const float* __restrict__ theta_log,
                           const float* __restrict__ gamma_log,
                           const float* __restrict__ b_re,
                           const float* __restrict__ c_re,
                           const float* __restrict__ c_im,
                           float* __restrict__ ws) {
  __shared__ float caRe[NO_][NX_], caIm[NO_][NX_];
  __shared__ float GRe[NO_][NO_], GIm[NO_][NO_];
  __shared__ float HiRe[NO_][NO_], HiIm[NO_][NO_];
  const int s = threadIdx.x;  // 0..127

  float lam = expf(-expf(nu_log[s]));
  float ph  = expf(theta_log[s]);
  float ar  = lam * cosf(ph);
  float ai  = lam * sinf(ph);
  ws[OFF_ARE + s] = ar;
  ws[OFF_AIM + s] = ai;

  float g = gamma_log[s];  // source bug: gamma_log itself, not exp()
#pragma unroll
  for (int i = 0; i < NU_; ++i) ws[OFF_BS + s * NU_ + i] = g * b_re[s * NU_ + i];

#pragma unroll
  for (int o = 0; o < NO_; ++o) {
    float cr = c_re[o * NX_ + s], ci = c_im[o * NX_ + s];
    caRe[o][s] = cr * ar - ci * ai;
    caIm[o][s] = 2.0f * cr * ai;  // cx_einsum bug replicated: 2*re(op1)*im(op2)
  }
  __syncthreads();

  if (s < 16) {  // G = ca * ca^H  (4x4 complex)
    int i = s >> 2, j = s & 3;
    float gr = 0.f, gi = 0.f;
    for (int k = 0; k < NX_; ++k) {
      gr += caRe[i][k] * caRe[j][k] + caIm[i][k] * caIm[j][k];
      gi += caIm[i][k] * caRe[j][k] - caRe[i][k] * caIm[j][k];
    }
    GRe[i][j] = gr; GIm[i][j] = gi;
  }
  __syncthreads();

  if (s == 0) {  // complex 4x4 Gauss-Jordan inverse
    float mr[4][8], mi[4][8];
    for (int i = 0; i < 4; ++i)
      for (int j = 0; j < 8; ++j) {
        mr[i][j] = (j < 4) ? GRe[i][j] : ((j - 4 == i) ? 1.f : 0.f);
        mi[i][j] = (j < 4) ? GIm[i][j] : 0.f;
      }
    for (int k = 0; k < 4; ++k) {
      float pr = mr[k][k], pi = mi[k][k];
      float inv = 1.f / (pr * pr + pi * pi);
      float ipr = pr * inv, ipi = -pi * inv;
      for (int j = 0; j < 8; ++j) {
        float tr = mr[k][j] * ipr - mi[k][j] * ipi;
        float ti = mr[k][j] * ipi + mi[k][j] * ipr;
        mr[k][j] = tr; mi[k][j] = ti;
      }
      for (int i = 0; i < 4; ++i)
        if (i != k) {
          float fr = mr[i][k], fi = mi[i][k];
          for (int j = 0; j < 8; ++j) {
            mr[i][j] -= fr * mr[k][j] - fi * mi[k][j];
            mi[i][j] -= fr * mi[k][j] + fi * mr[k][j];
          }
        }
    }
    for (int i = 0; i < 4; ++i)
      for (int j = 0; j < 4; ++j) { HiRe[i][j] = mr[i][4 + j]; HiIm[i][j] = mi[i][4 + j]; }
  }
  __syncthreads();

  float pv[NO_];
#pragma unroll
  for (int o = 0; o < NO_; ++o) {
    float acc = 0.f;
#pragma unroll
    for (int j = 0; j < NO_; ++j)
      acc += caRe[j][s] * HiRe[j][o] + caIm[j][s] * HiIm[j][o];  // Re(ca^H * Ginv)
    pv[o] = acc;
    ws[OFF_PIN + s * NO_ + o] = acc;
  }
  for (int b = 0; b < B_; ++b) {  // x0 = Re(pinv) @ y0  (real per cx_einsum bug)
    float x0 = 0.f;
#pragma unroll
    for (int o = 0; o < NO_; ++o) x0 += pv[o] * y0[b * NO_ + o];
    ws[OFF_X0 + b * NX_ + s] = x0;
  }
}

// ---------------------------------------------------------------------------
// Phase 2: propagate chunk carries. carry[c] = a^L * carry[c-1] + final[c-1].
// ---------------------------------------------------------------------------
__global__ void ldss_carry(float* __restrict__ ws) {
  int idx = blockIdx.x * blockDim.x + threadIdx.x;  // 0..B*NX-1
  int b = idx >> 7, s = idx & (NX_ - 1);
  float ar = ws[OFF_ARE + s], ai = ws[OFF_AIM + s];
#pragma unroll
  for (int k = 0; k < 8; ++k) {  // a^256 via 8 squarings
    float tr = ar * ar - ai * ai;
    ai = 2.f * ar * ai;
    ar = tr;
  }
  float er = 0.f, ei = 0.f;
  for (int c = 0; c < NC_; ++c) {
    int o = (b * NC_ + c) * NX_ + s;
    ws[OFF_CRE + o] = er;
    ws[OFF_CIM + o] = ei;
    float fr = ws[OFF_FRE + o], fi = ws[OFF_FIM + o];
    float nr = ar * er - ai * ei + fr;
    float ni = ai * er + ar * ei + fi;
    er = nr; ei = ni;
  }
}

// ---------------------------------------------------------------------------
// Phases 1 & 3: chunk-local complex scan. Phase 3 also does the output
// projection y = C_re*x_re - C_im*x_im + D*u with V_WMMA_F32_16X16X4_F32 on
// 16-timestep LDS tiles (stride-132 padding; all fragment loads = ds_load_b64,
// no predication). Block = 128 threads (thread = channel s, 4 wave32s).
// ---------------------------------------------------------------------------
template <int PHASE>
__global__ void ldss_scan(const float* __restrict__ u,
                          const float* __restrict__ c_re,
                          const float* __restrict__ c_im,
                          const float* __restrict__ dmat,
                          float* __restrict__ ws,
                          float* __restrict__ out) {
  extern __shared__ float smem[];
  float* uT     = smem + SM_UT;
  float* xReLds = smem + SM_XRE;
  float* xImLds = smem + SM_XIM;
  float* bReP   = smem + SM_BRE;  // zero-padded C_re, [16][XS_]
  float* bImP   = smem + SM_BIM;  // zero-padded -C_im, [16][XS_]
  float* dLds   = smem + SM_D;
  float* pc     = smem + SM_PC;

  const int b = blockIdx.x / NC_, c = blockIdx.x % NC_;
  const int tid = threadIdx.x, s = tid;
  const int lane = tid & 31, wave = tid >> 5;

  float ar = ws[OFF_ARE + s], ai = ws[OFF_AIM + s];
  float bs[NU_];
#pragma unroll
  for (int i = 0; i < NU_; ++i) bs[i] = ws[OFF_BS + s * NU_ + i];

  float xr, xi;
  if (PHASE == 1) {
    xr = 0.f; xi = 0.f;
  } else {
    int o = (b * NC_ + c) * NX_ + s;
    xr = ws[OFF_CRE + o]; xi = ws[OFF_CIM + o];
  }
  if (PHASE == 3) {
    // Stage zero-padded B matrices: rows 0..3 = C_re / -C_im, rows 4..15 = 0.
    for (int k = tid; k < 16 * NX_; k += 128) {
      int n = k >> 7, sc = k & (NX_ - 1);
      bReP[n * XS_ + sc] = (n < NO_) ? c_re[k] : 0.f;
      bImP[n * XS_ + sc] = (n < NO_) ? -c_im[k] : 0.f;
    }
    if (tid < NO_ * NU_) dLds[tid] = dmat[tid];
  }
  const float x0term = (c == 0) ? ar * ws[OFF_X0 + b * NX_ + s] : 0.f;
  const int t0 = c * L_;

  for (int tile = 0; tile < L_ / 16; ++tile) {
    const int tb = t0 + tile * 16;
    __syncthreads();
    uT[tid] = u[(size_t)(b * T_ + tb + (tid >> 3)) * NU_ + (tid & 7)];
    if (tile + 1 < L_ / 16)
      __builtin_prefetch(&u[(size_t)(b * T_ + tb + 16 + (tid >> 3)) * NU_ + (tid & 7)], 0, 1);
    __syncthreads();

#pragma unroll
    for (int tt = 0; tt < 16; ++tt) {
      float bu = 0.f;
#pragma unroll
      for (int i = 0; i < NU_; ++i) bu = fmaf(bs[i], uT[tt * NU_ + i], bu);
      if (c == 0 && tile == 0 && tt == 0) bu += x0term;  // first = B@u0 + aRe*x0
      float nr = fmaf(ar, xr, fmaf(-ai, xi, bu));
      float ni = fmaf(ai, xr, ar * xi);
      xr = nr; xi = ni;
      if (PHASE == 3) {
        xReLds[tt * XS_ + s] = xr;
        xImLds[tt * XS_ + s] = xi;
      }
    }

    if (PHASE == 3) {
      __syncthreads();
      v8f acc = {0.f, 0.f, 0.f, 0.f, 0.f, 0.f, 0.f, 0.f};
      const int M = lane & 15;            // A row (timestep)
      const int koff = (lane >> 4) << 1;  // A/B K half: 0 or 2
      const int n = lane & 15;            // B/D column (output channel, pad 4->16)
#pragma unroll
      for (int kk = 0; kk < 8; ++kk) {    // K sweep over this wave's 32 channels
        const int sb = wave * 32 + kk * 4;
#if HAS_WMMA_F32X4
        v2f af, bf;
        af.x = xReLds[M * XS_ + sb + koff];
        af.y = xReLds[M * XS_ + sb + koff + 1];
        bf.x = bReP[n * XS_ + sb + koff];
        bf.y = bReP[n * XS_ + sb + koff + 1];
        acc = __builtin_amdgcn_wmma_f32_16x16x4_f32(false, af, false, bf, (short)0, acc,
                                                    false, false);
        af.x = xImLds[M * XS_ + sb + koff];
        af.y = xImLds[M * XS_ + sb + koff + 1];
        bf.x = bImP[n * XS_ + sb + koff];
        bf.y = bImP[n * XS_ + sb + koff + 1];
        acc = __builtin_amdgcn_wmma_f32_16x16x4_f32(false, af, false, bf, (short)0, acc,
                                                    false, false);
#else
#pragma unroll
        for (int vg = 0; vg < 8; ++vg) {
          const int Mr = vg + ((lane >> 4) << 3);
          float a0 = 0.f;
#pragma unroll
          for (int k = 0; k < 4; ++k) {
            a0 += xReLds[Mr * XS_ + sb + k] * bReP[n * XS_ + sb + k] +
                  xImLds[Mr * XS_ + sb + k] * bImP[n * XS_ + sb + k];
          }
          acc[vg] += a0;
        }
#endif
      }
#pragma unroll
      for (int vg = 0; vg < 8; ++vg) pc[wave * 256 + vg * 32 + lane] = acc[vg];
      __syncthreads();
      if (tid < 64) {  // (tt, o) pairs: reduce 4 waves, add D@u, store
        const int tt = tid >> 2, o = tid & 3;
        const int vg = tt & 7, ln = ((tt >> 3) << 4) + o;  // C/D frag layout
        float y = pc[0 * 256 + vg * 32 + ln] + pc[1 * 256 + vg * 32 + ln] +
                  pc[2 * 256 + vg * 32 + ln] + pc[3 * 256 + vg * 32 + ln];
        float du = 0.f;
#pragma unroll
        for (int i = 0; i < NU_; ++i) du = fmaf(dLds[o * NU_ + i], uT[tt * NU_ + i], du);
        out[(size_t)(b * T_ + tb + tt) * NO_ + o] = y + du;
      }
    }
  }

  if (PHASE == 1) {
    int o = (b * NC_ + c) * NX_ + s;
    ws[OFF_FRE + o] = xr;
    ws[OFF_FIM + o] = xi;
  }
}

// ---------------------------------------------------------------------------
extern "C" void kernel_launch(void* const* d_in, const int* in_sizes, int n_in,
                              void* d_out, int out_size, void* d_ws, size_t ws_size,
                              hipStream_t stream) {
  (void)in_sizes; (void)n_in; (void)out_size; (void)ws_size;
  const float* u         = (const float*)d_in[0];
  const float* y0        = (const float*)d_in[1];
  const float* nu_log    = (const float*)d_in[2];
  const float* theta_log = (const float*)d_in[3];
  const float* gamma_log = (const float*)d_in[4];
  const float* b_re      = (const float*)d_in[5];
  // d_in[6] = b_im: dead under the source's cx_einsum bug (imag drive is 0)
  const float* c_re      = (const float*)d_in[7];
  const float* c_im      = (const float*)d_in[8];
  const float* dmat      = (const float*)d_in[9];
  float* out = (float*)d_out;
  float* ws  = (float*)d_ws;

  ldss_setup<<<1, 128, 0, stream>>>(y0, nu_log, theta_log, gamma_log, b_re, c_re, c_im, ws);
  ldss_scan<1><<<B_ * NC_, 128, SM_TOT1 * sizeof(float), stream>>>(u, c_re, c_im, dmat, ws, out);
  ldss_carry<<<(B_ * NX_) / 128, 128, 0, stream>>>(ws);
  ldss_scan<3><<<B_ * NC_, 128, SM_TOT3 * sizeof(float), stream>>>(u, c_re, c_im, dmat, ws, out);
}